// CRFLayer_83640193122949
// MI455X (gfx1250) — compile-verified
//
#include <hip/hip_runtime.h>
#include <hip/hip_bf16.h>
#include <math.h>

#define B_ 512
#define T_ 1024
#define K_ 48
#define MB 16      // batches per wave (WMMA M dimension)
#define WAVES 4    // waves per block
#define LPT 24     // emission loads per lane per step (MB*K_/32)

typedef __attribute__((ext_vector_type(2))) float v2f;
typedef __attribute__((ext_vector_type(8))) float v8f;

// ---------------------------------------------------------------------------
// Forward (log Z) pass: logsumexp recursion as f32 WMMA GEMM in the exp domain.
// One wave handles 16 batches; 3 N-tiles x 12 k-chunks of V_WMMA_F32_16X16X4_F32.
// B fragments (exp(transitions)) are re-loaded from LDS per WMMA (one
// ds_load_2addr each) to keep VGPR usage under 256 — avoids s_set_vgpr_msb
// overhead in the sequential T=1024 loop and allows wave co-residency.
// Emissions for step t+1 are prefetched into registers while step t computes.
// Hot-loop exp/log use native v_exp_f32/v_log_f32 (max-shift keeps args tame).
// ---------------------------------------------------------------------------
__global__ __launch_bounds__(32 * WAVES) void crf_forward_kernel(
    const float* __restrict__ em, const int* __restrict__ tags,
    const float* __restrict__ startT, const float* __restrict__ endT,
    const float* __restrict__ trans, float* __restrict__ logz)
{
    __shared__ float sExpT[K_ * K_];                 // exp(transitions), 9 KB
    __shared__ float sAlpha[WAVES][MB * K_];         // per-wave alpha
    __shared__ float sE[WAVES][MB * K_];             // staged emissions for step t
    __shared__ float sMp[WAVES][32];                 // partial maxima (2 halves)
    __shared__ float sM[WAVES][MB];                  // per-batch running max
    __shared__ int   sMk[WAVES][MB];                 // per-batch mask at step t

    const int tid  = threadIdx.x;
    const int lane = tid & 31;
    const int warp = tid >> 5;
    const int wid  = blockIdx.x * WAVES + warp;
    const int b0   = wid * MB;

    for (int i = tid; i < K_ * K_; i += blockDim.x) sExpT[i] = expf(trans[i]);
    __syncthreads();

    float* aw = sAlpha[warp];
    float* ew = sE[warp];

    // alpha(t=0) = start + emissions[:,0,:]   (t=0 always inside lengths)
    for (int i = lane; i < MB * K_; i += 32) {
        int b = i / K_, k = i % K_;
        aw[i] = startT[k] + em[(size_t)(b0 + b) * T_ * K_ + k];
    }
    __syncthreads();

    const int halfsel = (lane < 16) ? 0 : 2;   // A/B K-row pair select
    const int col16   = lane & 15;

    // software-pipeline prologue: emissions for t=1 into registers
    float pre[LPT];
    #pragma unroll
    for (int j = 0; j < LPT; ++j) {
        int i = lane + j * 32;
        int b = i / K_, k = i % K_;
        pre[j] = em[(size_t)(b0 + b) * T_ * K_ + (size_t)1 * K_ + k];
    }

    for (int t = 1; t < T_; ++t) {
        // spill prefetched emissions to LDS
        #pragma unroll
        for (int j = 0; j < LPT; ++j) ew[lane + j * 32] = pre[j];
        // per-batch running max: all 32 lanes scan 24 elems, then 2-way combine
        {
            const int bb = lane & 15, hh = lane >> 4;
            float mx = aw[bb * K_ + hh * 24];
            #pragma unroll
            for (int k = 1; k < 24; ++k) mx = fmaxf(mx, aw[bb * K_ + hh * 24 + k]);
            sMp[warp][lane] = mx;
        }
        if (lane < MB) {
            sMk[warp][lane] = (tags[(size_t)(b0 + lane) * T_ + t] > -1) ? 1 : 0;
            sM[warp][lane]  = fmaxf(sMp[warp][lane], sMp[warp][lane + 16]);
        }
        __syncthreads();

        // issue next step's emission loads early (latency hidden by compute)
        if (t + 1 < T_) {
            #pragma unroll
            for (int j = 0; j < LPT; ++j) {
                int i = lane + j * 32;
                int b = i / K_, k = i % K_;
                pre[j] = em[(size_t)(b0 + b) * T_ * K_ + (size_t)(t + 1) * K_ + k];
            }
        }

        // A fragments: exp(alpha - m), 16x4 f32 layout (args <= 0, native exp ok)
        const int   Mrow = col16;
        const float mrow = sM[warp][Mrow];
        v2f Af[12];
        #pragma unroll
        for (int kc = 0; kc < 12; ++kc) {
            int k0 = 4 * kc + halfsel;
            Af[kc][0] = __expf(aw[Mrow * K_ + k0]     - mrow);
            Af[kc][1] = __expf(aw[Mrow * K_ + k0 + 1] - mrow);
        }

        // D(16x16) = expA(16x48) x expT(48x16-tile), 12 chained k=4 WMMAs;
        // B fragments streamed from LDS (ds_load_2addr, offset1 = one row)
        v8f acc[3];
        #pragma unroll
        for (int nt = 0; nt < 3; ++nt) {
            v8f c = {};
            #pragma unroll
            for (int kc = 0; kc < 12; ++kc) {
                const int r = 4 * kc + halfsel;
                const int cc = nt * 16 + col16;
                v2f Bfrag;
                Bfrag[0] = sExpT[r * K_ + cc];
                Bfrag[1] = sExpT[(r + 1) * K_ + cc];
                c = __builtin_amdgcn_wmma_f32_16x16x4_f32(
                        false, Af[kc], false, Bfrag, (short)0, c, false, false);
            }
            acc[nt] = c;
        }

        // writeback: alpha'[M][N] = m[M] + log(D) + e[M][N], masked
        #pragma unroll
        for (int nt = 0; nt < 3; ++nt) {
            int N = nt * 16 + col16;
            #pragma unroll
            for (int v = 0; v < 8; ++v) {
                int   M  = (lane < 16) ? v : (8 + v);
                int   mk = sMk[warp][M];
                float nv = sM[warp][M] + __logf(acc[nt][v]) + ew[M * K_ + N];
                float ov = aw[M * K_ + N];
                aw[M * K_ + N] = mk ? nv : ov;
            }
        }
        __syncthreads();   // alpha/emission buffers safe to reuse next iteration
    }

    if (lane < MB) {
        int b = lane;
        float mx = -3.4e38f;
        for (int k = 0; k < K_; ++k) mx = fmaxf(mx, aw[b * K_ + k] + endT[k]);
        float s = 0.f;
        for (int k = 0; k < K_; ++k) s += expf(aw[b * K_ + k] + endT[k] - mx);
        logz[b0 + b] = mx + logf(s);
    }
}

// ---------------------------------------------------------------------------
// Numerator (gold path) score: one thread per batch
// ---------------------------------------------------------------------------
__global__ void crf_score_kernel(
    const float* __restrict__ em, const int* __restrict__ tags,
    const float* __restrict__ startT, const float* __restrict__ endT,
    const float* __restrict__ trans, float* __restrict__ score)
{
    int b = blockIdx.x * blockDim.x + threadIdx.x;
    if (b >= B_) return;
    int prev = tags[(size_t)b * T_];
    float sc = startT[prev] + em[(size_t)b * T_ * K_ + prev];
    for (int t = 1; t < T_; ++t) {
        int tg = tags[(size_t)b * T_ + t];
        if (tg > -1) {
            sc += trans[prev * K_ + tg] + em[(size_t)b * T_ * K_ + (size_t)t * K_ + tg];
            prev = tg;
        }
    }
    sc += endT[prev];
    score[b] = sc;
}

// ---------------------------------------------------------------------------
// Viterbi: one block per batch; transitions + all backpointers in LDS (57 KB,
// comfortably inside CDNA5's 320 KB WGP LDS) so the backtrace is LDS-latency
// instead of 1024 dependent global loads.
// ---------------------------------------------------------------------------
__global__ __launch_bounds__(64) void crf_viterbi_kernel(
    const float* __restrict__ em, const int* __restrict__ tags,
    const float* __restrict__ startT, const float* __restrict__ endT,
    const float* __restrict__ trans, float* __restrict__ outTags)
{
    __shared__ float sTrans[K_ * K_];          // 9 KB
    __shared__ float va[K_];
    __shared__ unsigned char bp[T_ * K_];      // 48 KB backpointers

    const int b   = blockIdx.x;
    const int tid = threadIdx.x;

    for (int i = tid; i < K_ * K_; i += 64) sTrans[i] = trans[i];
    if (tid < K_) va[tid] = startT[tid] + em[(size_t)b * T_ * K_ + tid];
    __syncthreads();

    for (int t = 1; t < T_; ++t) {
        int mk = tags[(size_t)b * T_ + t] > -1;
        float nv = 0.f;
        if (tid < K_) {
            float e = em[(size_t)b * T_ * K_ + (size_t)t * K_ + tid];
            float best = -3.4e38f;
            int bi = 0;
            for (int i = 0; i < K_; ++i) {
                float s = va[i] + sTrans[i * K_ + tid];
                if (s > best) { best = s; bi = i; }   // first max on ties (argmax)
            }
            nv = mk ? (best + e) : va[tid];
            bp[t * K_ + tid] = (unsigned char)(mk ? bi : tid);
        }
        __syncthreads();
        if (tid < K_) va[tid] = nv;
        __syncthreads();
    }

    if (tid == 0) {
        float best = -3.4e38f;
        int cur = 0;
        for (int j = 0; j < K_; ++j) {
            float s = va[j] + endT[j];
            if (s > best) { best = s; cur = j; }
        }
        for (int t = T_ - 1; t >= 1; --t) {
            int mk = tags[(size_t)b * T_ + t] > -1;
            outTags[(size_t)b * T_ + t] = mk ? (float)cur : -1.0f;
            cur = bp[t * K_ + cur];
        }
        outTags[(size_t)b * T_] = (float)cur;   // t=0 always valid (len >= 1)
    }
}

// ---------------------------------------------------------------------------
// Final reduction: loss = sum_b (logZ[b] - score[b])
// ---------------------------------------------------------------------------
__global__ __launch_bounds__(256) void crf_loss_reduce(
    const float* __restrict__ logz, const float* __restrict__ score, float* __restrict__ out)
{
    __shared__ float red[256];
    float s = 0.f;
    for (int b = threadIdx.x; b < B_; b += 256) s += logz[b] - score[b];
    red[threadIdx.x] = s;
    __syncthreads();
    for (int off = 128; off > 0; off >>= 1) {
        if (threadIdx.x < off) red[threadIdx.x] += red[threadIdx.x + off];
        __syncthreads();
    }
    if (threadIdx.x == 0) out[0] = red[0];
}

extern "C" void kernel_launch(void* const* d_in, const int* in_sizes, int n_in,
                              void* d_out, int out_size, void* d_ws, size_t ws_size,
                              hipStream_t stream) {
    const float* em     = (const float*)d_in[0];
    const int*   tags   = (const int*)d_in[1];
    const float* startT = (const float*)d_in[2];
    const float* endT   = (const float*)d_in[3];
    const float* trans  = (const float*)d_in[4];
    float* out = (float*)d_out;

    float* logz  = (float*)d_ws;       // B_ floats
    float* score = logz + B_;          // B_ floats

    crf_forward_kernel<<<B_ / (MB * WAVES), 32 * WAVES, 0, stream>>>(
        em, tags, startT, endT, trans, logz);
    crf_score_kernel<<<(B_ + 255) / 256, 256, 0, stream>>>(
        em, tags, startT, endT, trans, score);
    crf_viterbi_kernel<<<B_, 64, 0, stream>>>(
        em, tags, startT, endT, trans, out + 1);
    crf_loss_reduce<<<1, 256, 0, stream>>>(logz, score, out);
}